// DMoN3P_17188459118720
// MI455X (gfx1250) — compile-verified
//
#include <hip/hip_runtime.h>
#include <math.h>
#include <stdint.h>

#define KDIM 64

typedef __attribute__((ext_vector_type(2))) float v2f;
typedef __attribute__((ext_vector_type(8))) float v8f;
typedef __attribute__((ext_vector_type(4))) unsigned int u32x4;
typedef __attribute__((ext_vector_type(8))) unsigned int u32x8;

__device__ __forceinline__ v8f wmma_f32_16x16x4(v2f a, v2f b, v8f c) {
  // D = A(16x4 f32) x B(4x16 f32) + C(16x16 f32)
  return __builtin_amdgcn_wmma_f32_16x16x4_f32(false, a, false, b, (short)0, c, false, false);
}

// ---- TDM: DMA a contiguous 64x64 f32 tile from global memory into LDS ----
// D# per CDNA5 ISA: group0 = {count/flags, lds_addr,
// global_addr[56:0], type=2}; group1 = {data_size=4B, tensor_dim0/1=64,
// tile_dim0/1=64, tensor_dim0_stride=64}. 2D tensor -> 2-operand asm form
// (VADDR2/3 = NULL). Completion tracked with TENSORcnt.
__device__ __forceinline__ void tdm_load_tile_64x64_f32(const float* gsrc, unsigned lds_byte_off) {
  unsigned long long ga = (unsigned long long)(uintptr_t)gsrc;
  u32x4 g0;
  g0[0] = 1u;                                                  // count=1 (valid user D#)
  g0[1] = lds_byte_off;                                        // lds_addr (bytes)
  g0[2] = (unsigned)ga;                                        // global_addr[31:0]
  g0[3] = (unsigned)((ga >> 32) & 0x01FFFFFFu) | 0x80000000u;  // global_addr[56:32] | type=2<<30
  u32x8 g1;
  g1[0] = 0x00020000u;   // workgroup_mask=0, data_size=2 (4 bytes)
  g1[1] = 64u << 16;     // tensor_dim0 = 64 (bits [79:48] low half)
  g1[2] = 64u << 16;     // tensor_dim1 = 64 (bits [111:80] low half)
  g1[3] = 64u << 16;     // tile_dim0   = 64 (bits [127:112])
  g1[4] = 64u;           // tile_dim1   = 64 (bits [143:128])
  g1[5] = 64u;           // tensor_dim0_stride = 64 (bits [207:160] low half)
  g1[6] = 0u;
  g1[7] = 0u;
  asm volatile("tensor_load_to_lds %0, %1" :: "s"(g0), "s"(g1) : "memory");
}

// ---------------- zero scratch ----------------
__global__ void zero_kernel(float* __restrict__ p, size_t n) {
  size_t i = (size_t)blockIdx.x * blockDim.x + threadIdx.x;
  size_t stride = (size_t)gridDim.x * blockDim.x;
  for (; i < n; i += stride) p[i] = 0.0f;
}

// ---------------- row softmax + column sums + neg-entropy ----------------
__global__ void __launch_bounds__(64) row_softmax_kernel(const float* __restrict__ logits,
    float* __restrict__ S, float* __restrict__ colsum, float* __restrict__ ent_acc, int nrows) {
  int row = blockIdx.x;
  int c = threadIdx.x;
  __shared__ float red[64];
  float v = logits[(size_t)row * KDIM + c];
  red[c] = v;
  __syncthreads();
  for (int s = 32; s > 0; s >>= 1) {
    if (c < s) red[c] = fmaxf(red[c], red[c + s]);
    __syncthreads();
  }
  float mx = red[0];
  __syncthreads();
  float e = __expf(v - mx);
  red[c] = e;
  __syncthreads();
  for (int s = 32; s > 0; s >>= 1) {
    if (c < s) red[c] += red[c + s];
    __syncthreads();
  }
  float p = e / red[0];
  __syncthreads();
  S[(size_t)row * KDIM + c] = p;
  atomicAdd(&colsum[c], p);
  red[c] = p * __logf(p + 1e-9f);
  __syncthreads();
  for (int s = 32; s > 0; s >>= 1) {
    if (c < s) red[c] += red[c + s];
    __syncthreads();
  }
  if (c == 0) atomicAdd(ent_acc, red[0]);
}

// ---------------- edge degree: deg[dst[e]] += w[e] ----------------
__global__ void edge_degree_kernel(const int* __restrict__ dst, const float* __restrict__ w,
                                   float* __restrict__ deg, int n) {
  int e = blockIdx.x * blockDim.x + threadIdx.x;
  if (e < n) atomicAdd(&deg[dst[e]], w[e]);
}

// ---------------- edge scatter: acc[dst[e], :] += w[e] * S[src[e], :] ----------------
// 16 lanes per edge, float4 (b128) gather per lane.
__global__ void __launch_bounds__(256) edge_scatter_kernel(const float* __restrict__ S,
    const int* __restrict__ src, const int* __restrict__ dst, const float* __restrict__ w,
    float* __restrict__ acc, int n) {
  int e = blockIdx.x * 16 + (threadIdx.x >> 4);
  int c4 = (threadIdx.x & 15) * 4;
  if (e < n) {
    float wv = w[e];
    int i = src[e];
    int j = dst[e];
    const float4 sv = *(const float4*)&S[(size_t)i * KDIM + c4];
    float* dp = &acc[(size_t)j * KDIM + c4];
    atomicAdd(dp + 0, wv * sv.x);
    atomicAdd(dp + 1, wv * sv.y);
    atomicAdd(dp + 2, wv * sv.z);
    atomicAdd(dp + 3, wv * sv.w);
  }
}

// ---------------- omega scaling of A,C and Mnorm accumulation ----------------
__global__ void __launch_bounds__(64) omega_scale_kernel(float* __restrict__ A, float* __restrict__ Cc,
    const float* __restrict__ degX, const float* __restrict__ degZ, float* __restrict__ scal, int Y) {
  int y = blockIdx.x;
  int c = threadIdx.x;
  __shared__ float om;
  if (c == 0) {
    float dx = degX[y];
    float dz = degZ[y];
    bool valid = (dx > 0.0f) && (dz > 0.0f);
    float prod = dx * dz;
    om = valid ? 1.0f / (prod + 1e-9f) : 0.0f;
    if (valid) atomicAdd(&scal[0], prod);
  }
  __syncthreads();
  size_t idx = (size_t)y * KDIM + c;
  A[idx] *= om;
  Cc[idx] *= om;
}

// ---------------- R[64x64] += X^T * Yd over a chunk of rows (f32 WMMA) ----------------
__global__ void __launch_bounds__(32) gemm_atb_wmma_kernel(const float* __restrict__ X,
    const float* __restrict__ Yd, float* __restrict__ R, int nrows, int chunk) {
  int tile = blockIdx.x & 15;        // 4x4 tiles of 16x16 -> 64x64 result
  int ck = blockIdx.x >> 4;
  int l0 = (tile >> 2) * 16;
  int m0 = (tile & 3) * 16;
  int y0 = ck * chunk;
  int y1 = y0 + chunk;
  if (y1 > nrows) y1 = nrows;
  int lane = threadIdx.x;
  int half = lane >> 4;              // 0 | 1
  int lm = lane & 15;
  v8f acc = {};
#pragma unroll 4
  for (int y = y0; y < y1; y += 4) {
    size_t ra = (size_t)(y + 2 * half) * KDIM;
    size_t rb = (size_t)(y + 2 * half + 1) * KDIM;
    v2f a, b;
    a.x = X[ra + l0 + lm];           // A[m=l, k=y]  (16x4 tile of X^T)
    a.y = X[rb + l0 + lm];
    b.x = Yd[ra + m0 + lm];          // B[k=y, n=m]  (4x16 tile of Yd)
    b.y = Yd[rb + m0 + lm];
    acc = wmma_f32_16x16x4(a, b, acc);
  }
  for (int v = 0; v < 8; v++) {
    atomicAdd(&R[(size_t)(l0 + v + 8 * half) * KDIM + m0 + lm], acc[v]);
  }
}

// ---------------- E scaling, alpha/gamma softmax, Q_exp ----------------
__global__ void __launch_bounds__(64) finalize_e_kernel(const float* __restrict__ ExyRaw,
    const float* __restrict__ EyzRaw, float* __restrict__ alpha, float* __restrict__ gam,
    float* __restrict__ scal) {
  __shared__ float EXY[4096];
  __shared__ float EYZ[4096];
  __shared__ float aX[64], aY[64], aZ[64], red[64];
  int t = threadIdx.x;
  float inv = 1.0f / (scal[0] + 1e-9f);
  for (int i = t; i < 4096; i += 64) {
    EXY[i] = ExyRaw[i] * inv;
    EYZ[i] = EyzRaw[i] * inv;
  }
  __syncthreads();
  float sx = 0.0f, sy = 0.0f, sz = 0.0f;
  for (int i = 0; i < 64; i++) {
    sx += EXY[t * 64 + i];   // aX: row sums of E_XY
    sy += EXY[i * 64 + t];   // aY: col sums of E_XY
    sz += EYZ[i * 64 + t];   // aZ: col sums of E_YZ
  }
  aX[t] = sx; aY[t] = sy; aZ[t] = sz;
  __syncthreads();
  // alpha = softmax(3*E_XY, axis=0): thread t owns column t
  float mxv = -3.0e38f;
  for (int l = 0; l < 64; l++) mxv = fmaxf(mxv, 3.0f * EXY[l * 64 + t]);
  float asum = 0.0f;
  for (int l = 0; l < 64; l++) {
    float e = __expf(3.0f * EXY[l * 64 + t] - mxv);
    EXY[l * 64 + t] = e;
    asum += e;
  }
  for (int l = 0; l < 64; l++) {
    float av = EXY[l * 64 + t] / asum;
    EXY[l * 64 + t] = av;
    alpha[l * 64 + t] = av;
  }
  // gamma = softmax(3*E_YZ, axis=1): thread t owns row t
  float mgv = -3.0e38f;
  for (int n = 0; n < 64; n++) mgv = fmaxf(mgv, 3.0f * EYZ[t * 64 + n]);
  float gsum = 0.0f;
  for (int n = 0; n < 64; n++) {
    float e = __expf(3.0f * EYZ[t * 64 + n] - mgv);
    EYZ[t * 64 + n] = e;
    gsum += e;
  }
  for (int n = 0; n < 64; n++) {
    float gv = EYZ[t * 64 + n] / gsum;
    EYZ[t * 64 + n] = gv;
    gam[t * 64 + n] = gv;
  }
  // termL[t] = sum_l alpha[l,t]*aX[l]; termR[t] = sum_n gamma[t,n]*aZ[n]
  float tl = 0.0f;
  for (int l = 0; l < 64; l++) tl += EXY[l * 64 + t] * aX[l];
  float tr = 0.0f;
  for (int n = 0; n < 64; n++) tr += EYZ[t * 64 + n] * aZ[n];
  red[t] = tl * aY[t] * tr;
  __syncthreads();
  for (int s = 32; s > 0; s >>= 1) {
    if (t < s) red[t] += red[t + s];
    __syncthreads();
  }
  if (t == 0) scal[5] = red[0] * inv;   // Q_exp
}

// ---------------- fused left = AY@alpha, right = CY@gamma^T, Q_obs raw sum ----------------
// alpha/gamma staged into LDS via TDM descriptor DMA; B operands read from LDS.
__global__ void __launch_bounds__(32) qobs_wmma_kernel(const float* __restrict__ AY,
    const float* __restrict__ CY, const float* __restrict__ SY,
    const float* __restrict__ alpha, const float* __restrict__ gam,
    float* __restrict__ scal) {
  __shared__ float lds_ag[8192];   // [0..4095] alpha, [4096..8191] gamma
  int y0 = blockIdx.x * 16;
  int lane = threadIdx.x;
  int half = lane >> 4;
  int lm = lane & 15;

  // flat LDS address: aperture in [63:32], byte offset in [31:0]
  unsigned ldsbase = (unsigned)(uintptr_t)(void*)lds_ag;
  tdm_load_tile_64x64_f32(alpha, ldsbase);
  tdm_load_tile_64x64_f32(gam, ldsbase + 64u * 64u * 4u);
  __builtin_amdgcn_s_wait_tensorcnt(0);

  v8f left[4] = {};
  v8f right[4] = {};
  // left[y, m] = sum_l AY[y,l] * alpha[l,m]
  for (int l0 = 0; l0 < 64; l0 += 4) {
    v2f a = *(const v2f*)&AY[(size_t)(y0 + lm) * KDIM + l0 + 2 * half];  // b64 load
    for (int tt = 0; tt < 4; tt++) {
      int m0 = tt * 16;
      v2f b;
      b.x = lds_ag[(l0 + 2 * half) * 64 + m0 + lm];
      b.y = lds_ag[(l0 + 2 * half + 1) * 64 + m0 + lm];
      left[tt] = wmma_f32_16x16x4(a, b, left[tt]);
    }
  }
  // right[y, m] = sum_n CY[y,n] * gamma[m,n]
  for (int n0 = 0; n0 < 64; n0 += 4) {
    v2f a = *(const v2f*)&CY[(size_t)(y0 + lm) * KDIM + n0 + 2 * half];  // b64 load
    for (int tt = 0; tt < 4; tt++) {
      int m0 = tt * 16;
      v2f b = *(const v2f*)&lds_ag[4096 + (m0 + lm) * 64 + n0 + 2 * half]; // ds_load_b64
      right[tt] = wmma_f32_16x16x4(a, b, right[tt]);
    }
  }
  // triple product with S_Y, reduce within wave
  float part = 0.0f;
  for (int tt = 0; tt < 4; tt++) {
    for (int v = 0; v < 8; v++) {
      int yy = y0 + v + 8 * half;   // D row -> y
      int mm = tt * 16 + lm;        // D col -> m
      part += left[tt][v] * SY[(size_t)yy * KDIM + mm] * right[tt][v];
    }
  }
  for (int off = 16; off > 0; off >>= 1) part += __shfl_down(part, off, 32);
  if (lane == 0) atomicAdd(&scal[4], part);   // Q_obs raw (divide by Mnorm at the end)
}

// ---------------- final scalar combine ----------------
__global__ void finalize_loss_kernel(const float* __restrict__ csX, const float* __restrict__ csY,
    const float* __restrict__ csZ, const float* __restrict__ scal, float* __restrict__ out,
    int nX, int nY, int nZ) {
  float Mn = scal[0] + 1e-9f;
  float Qobs = scal[4] / Mn;
  float Qexp = scal[5];
  float Q = Qobs - 1.0f * Qexp;   // GAMMA = 1.0
  float fx = 0.0f, fy = 0.0f, fz = 0.0f;
  for (int i = 0; i < 64; i++) {
    fx += csX[i] * csX[i];
    fy += csY[i] * csY[i];
    fz += csZ[i] * csZ[i];
  }
  // sqrt(K)/n * ||colsum|| - 1, K = 64
  float regc = 1e-4f * ((8.0f / (float)nX) * sqrtf(fx) - 1.0f)
             + 1e-4f * ((8.0f / (float)nY) * sqrtf(fy) - 1.0f)
             + 1e-4f * ((8.0f / (float)nZ) * sqrtf(fz) - 1.0f);
  float rege = 1e-3f * (scal[1] / (float)nX + scal[2] / (float)nY + scal[3] / (float)nZ);
  out[0] = -Q + regc + rege;
}

extern "C" void kernel_launch(void* const* d_in, const int* in_sizes, int n_in,
                              void* d_out, int out_size, void* d_ws, size_t ws_size,
                              hipStream_t stream) {
  const float* SxL = (const float*)d_in[0];
  const float* SyL = (const float*)d_in[1];
  const float* SzL = (const float*)d_in[2];
  const int*   eixy = (const int*)d_in[3];   // [2,E]: row0 = i (X), row1 = j (Y)
  const float* wxy = (const float*)d_in[4];
  const int*   eiyz = (const int*)d_in[5];   // [2,E]: row0 = j (Y), row1 = k (Z)
  const float* wyz = (const float*)d_in[6];

  int nX = in_sizes[0] / KDIM;
  int nY = in_sizes[1] / KDIM;
  int nZ = in_sizes[2] / KDIM;
  int Exy = in_sizes[4];
  int Eyz = in_sizes[6];

  float* ws = (float*)d_ws;
  size_t nYK = (size_t)nY * KDIM;
  // ---- zero-initialized region (contiguous) ----
  float* A    = ws;                 // [nY,64] -> becomes AY after omega scale
  float* C    = A + nYK;            // [nY,64] -> becomes CY
  float* degX = C + nYK;            // [nY]
  float* degZ = degX + nY;          // [nY]
  float* ExyR = degZ + nY;          // [64,64] raw (pre 1/Mnorm)
  float* EyzR = ExyR + 4096;        // [64,64] raw
  float* csX  = EyzR + 4096;        // [64] column sums of S_X
  float* csY  = csX + 64;
  float* csZ  = csY + 64;
  float* scal = csZ + 64;           // [0]=Mnorm_acc [1..3]=ent [4]=Qobs_raw [5]=Qexp
  size_t zeroCount = (size_t)((scal + 8) - ws);
  // ---- no-init region ----
  float* SX    = scal + 8;                        // softmax(Sx)
  float* SYm   = SX + (size_t)nX * KDIM;          // softmax(Sy)
  float* SZ    = SYm + (size_t)nY * KDIM;         // softmax(Sz)
  float* alpha = SZ + (size_t)nZ * KDIM;          // [64,64]
  float* gam   = alpha + 4096;                    // [64,64]
  (void)ws_size; (void)n_in; (void)out_size;

  zero_kernel<<<2048, 256, 0, stream>>>(ws, zeroCount);

  row_softmax_kernel<<<nX, 64, 0, stream>>>(SxL, SX, csX, &scal[1], nX);
  row_softmax_kernel<<<nY, 64, 0, stream>>>(SyL, SYm, csY, &scal[2], nY);
  row_softmax_kernel<<<nZ, 64, 0, stream>>>(SzL, SZ, csZ, &scal[3], nZ);

  edge_degree_kernel<<<(Exy + 255) / 256, 256, 0, stream>>>(eixy + Exy, wxy, degX, Exy);
  edge_degree_kernel<<<(Eyz + 255) / 256, 256, 0, stream>>>(eiyz, wyz, degZ, Eyz);

  edge_scatter_kernel<<<(Exy + 15) / 16, 256, 0, stream>>>(SX, eixy, eixy + Exy, wxy, A, Exy);
  edge_scatter_kernel<<<(Eyz + 15) / 16, 256, 0, stream>>>(SZ, eiyz + Eyz, eiyz, wyz, C, Eyz);

  omega_scale_kernel<<<nY, 64, 0, stream>>>(A, C, degX, degZ, scal, nY);

  const int CHUNK = 512;             // multiple of 4; nY % 4 == 0
  int nchunks = (nY + CHUNK - 1) / CHUNK;
  gemm_atb_wmma_kernel<<<16 * nchunks, 32, 0, stream>>>(A, SYm, ExyR, nY, CHUNK);   // E_XY raw
  gemm_atb_wmma_kernel<<<16 * nchunks, 32, 0, stream>>>(SYm, C, EyzR, nY, CHUNK);   // E_YZ raw

  finalize_e_kernel<<<1, 64, 0, stream>>>(ExyR, EyzR, alpha, gam, scal);

  qobs_wmma_kernel<<<nY / 16, 32, 0, stream>>>(A, C, SYm, alpha, gam, scal);

  finalize_loss_kernel<<<1, 1, 0, stream>>>(csX, csY, csZ, scal, (float*)d_out, nX, nY, nZ);
}